// GraphCommunicationNetwork_46385646797013
// MI455X (gfx1250) — compile-verified
//
#include <hip/hip_runtime.h>
#include <hip/hip_bf16.h>
#include <math.h>

typedef __attribute__((ext_vector_type(2))) float v2f;
typedef __attribute__((ext_vector_type(8))) float v8f;

#define NAGT 1024
#define HID  64
#define NEDGE (NAGT * (NAGT - 1))

// ---------------------------------------------------------------------------
// Kernel 1: [A | B] = emb(1024x64) @ Wcat(64x128) via V_WMMA_F32_16X16X4_F32.
//   Wcat[k][n] = (n < 64) ? W1[k][n] : W1[64+k][n-64]   (W1 stored [128,64] RM)
//   A gets b1 folded in at store time.
// One wave (32 threads) per 16x16 output tile; 16 K-steps of K=4 fp32 WMMA.
// ---------------------------------------------------------------------------
__global__ void __launch_bounds__(32)
gcn_precompute_wmma(const float* __restrict__ emb,
                    const float* __restrict__ W1,
                    const float* __restrict__ b1,
                    float* __restrict__ Aout,   // [1024,64], includes b1
                    float* __restrict__ Bout)   // [1024,64]
{
    const int tile  = blockIdx.x;          // 0..511
    const int tileM = (tile >> 3) << 4;    // 64 M-tiles
    const int tileN = (tile & 7)  << 4;    // 8 N-tiles over 128 cols
    const int lane  = threadIdx.x;
    const int half  = lane >> 4;           // 0: lanes 0-15, 1: lanes 16-31
    const int l     = lane & 15;

    const bool isA  = (tileN < 64);
    const int  ncol = tileN + l;           // column in Wcat space
    const int  wcol = isA ? ncol : (ncol - 64);
    const int  wrow0 = isA ? 0 : 64;       // row offset into W1 for this half

    v8f c = {};
    #pragma unroll
    for (int kk = 0; kk < 16; ++kk) {
        const int ka = kk * 4 + 2 * half;  // this lane-half's K pair base
        v2f a, b;
        // A-frag (16x4 fp32): lanes 0-15 hold K=ka,ka+1 ; lanes 16-31 K=ka,ka+1 (+2)
        a.x = emb[(tileM + l) * HID + ka];
        a.y = emb[(tileM + l) * HID + ka + 1];
        // B-frag (4x16 fp32): row K striped across lanes within a VGPR
        b.x = W1[(wrow0 + ka)     * HID + wcol];
        b.y = W1[(wrow0 + ka + 1) * HID + wcol];
        c = __builtin_amdgcn_wmma_f32_16x16x4_f32(
                /*neg_a=*/false, a, /*neg_b=*/false, b,
                /*c_mod=*/(short)0, c, /*reuse_a=*/false, /*reuse_b=*/false);
    }

    // C/D layout: VGPR r -> M = r + 8*half, N = l
    const int rowbase = tileM + 8 * half;
    if (isA) {
        const float bias = b1[wcol];
        #pragma unroll
        for (int r = 0; r < 8; ++r)
            Aout[(rowbase + r) * HID + wcol] = c[r] + bias;
    } else {
        #pragma unroll
        for (int r = 0; r < 8; ++r)
            Bout[(rowbase + r) * HID + wcol] = c[r];
    }
}

// ---------------------------------------------------------------------------
// Kernel 2: per-edge  w = sigmoid( sum_k relu(A[i,k] + B[j,k]) * W2[k] + b2 )
// One block per i; A[i] (already has b1) and W2 staged in LDS; each thread
// streams B[j] rows as float4. Writes (edge_index_row, edge_index_col, weight)
// in the reference's compacted row-major order: e = i*1023 + j - (j>i).
// ---------------------------------------------------------------------------
__global__ void __launch_bounds__(256)
gcn_edge_kernel(const float* __restrict__ A,
                const float* __restrict__ B,
                const float* __restrict__ W2,
                const float* __restrict__ b2,
                float* __restrict__ out)
{
    __shared__ float sAi[HID];
    __shared__ float sW2[HID];
    const int i   = blockIdx.x;
    const int tid = threadIdx.x;

    if (tid < HID)             sAi[tid]       = A[i * HID + tid];
    else if (tid < 2 * HID)    sW2[tid - HID] = W2[tid - HID];
    __syncthreads();

    const float bias2 = b2[0];

    for (int j = tid; j < NAGT; j += 256) {
        if (j == i) continue;
        const float4* brow = (const float4*)(B + j * HID);
        float s = bias2;
        #pragma unroll
        for (int q = 0; q < 16; ++q) {
            const float4 bv = brow[q];
            s += fmaxf(sAi[4 * q + 0] + bv.x, 0.0f) * sW2[4 * q + 0];
            s += fmaxf(sAi[4 * q + 1] + bv.y, 0.0f) * sW2[4 * q + 1];
            s += fmaxf(sAi[4 * q + 2] + bv.z, 0.0f) * sW2[4 * q + 2];
            s += fmaxf(sAi[4 * q + 3] + bv.w, 0.0f) * sW2[4 * q + 3];
        }
        const float w = 1.0f / (1.0f + __expf(-s));
        const int e = i * (NAGT - 1) + j - (j > i ? 1 : 0);
        out[e]             = (float)i;   // edge_index row 0
        out[NEDGE + e]     = (float)j;   // edge_index row 1
        out[2 * NEDGE + e] = w;          // edge weight
    }
}

// ---------------------------------------------------------------------------
// Launch
// Inputs (setup_inputs order): node_embeddings[1024*64] f32, W1[128*64] f32,
// b1[64] f32, W2[64] f32, b2[1] f32.
// Workspace: A (256KB) + B (256KB) = 512KB.
// ---------------------------------------------------------------------------
extern "C" void kernel_launch(void* const* d_in, const int* in_sizes, int n_in,
                              void* d_out, int out_size, void* d_ws, size_t ws_size,
                              hipStream_t stream) {
    const float* emb = (const float*)d_in[0];
    const float* W1  = (const float*)d_in[1];
    const float* b1  = (const float*)d_in[2];
    const float* W2  = (const float*)d_in[3];
    const float* b2  = (const float*)d_in[4];
    float* out = (float*)d_out;

    float* wsA = (float*)d_ws;                 // [1024,64]
    float* wsB = wsA + (size_t)NAGT * HID;     // [1024,64]

    // 512 tiles (64 M-tiles x 8 N-tiles), one wave each
    gcn_precompute_wmma<<<dim3(512), dim3(32), 0, stream>>>(emb, W1, b1, wsA, wsB);
    // one block per source agent i
    gcn_edge_kernel<<<dim3(NAGT), dim3(256), 0, stream>>>(wsA, wsB, W2, b2, out);
}